// MaterialEmbeddingLayer_74217034875538
// MI455X (gfx1250) — compile-verified
//
#include <hip/hip_runtime.h>
#include <hip/hip_bf16.h>
#include <math.h>

#define N_MATN 50000
#define N_EDGE 800000
#define D_OPF  48
#define D_EMB  64

typedef __attribute__((ext_vector_type(2))) float v2f;
typedef __attribute__((ext_vector_type(8))) float v8f;

// ---- order-preserving float<->uint key for atomic max over floats ----
__device__ __forceinline__ unsigned fkey(float x) {
    unsigned u = __float_as_uint(x);
    return (u & 0x80000000u) ? ~u : (u | 0x80000000u);
}
__device__ __forceinline__ float fkey_inv(unsigned k) {
    unsigned u = (k & 0x80000000u) ? (k & 0x7FFFFFFFu) : ~k;
    return __uint_as_float(u);
}
__device__ __forceinline__ float lrelu(float x) { return x > 0.f ? x : 0.2f * x; }

// ---------------- init: zero out + scores, reset scalars ----------------
__global__ void init_ws_kernel(float* __restrict__ out, float* __restrict__ scores,
                               unsigned* __restrict__ maxkey, float* __restrict__ denom) {
    int i = blockIdx.x * blockDim.x + threadIdx.x;
    if (i < N_MATN * D_EMB) out[i] = 0.f;
    if (i < N_MATN + N_EDGE) scores[i] = 0.f;
    if (i == 0) { *maxkey = 0u; *denom = 0.f; }
}

// ---------------- h = materials @ W_mat^T (WMMA f32 16x16x4) ----------------
__global__ void h_gemm_kernel(const float* __restrict__ mat, const float* __restrict__ Wmat,
                              float* __restrict__ h) {
    const int wave = (blockIdx.x * blockDim.x + threadIdx.x) >> 5;
    const int lane = threadIdx.x & 31;
    const int NT = D_EMB / 16;                      // 4
    const int total = (N_MATN / 16) * NT;           // 12500
    if (wave >= total) return;                      // wave-uniform
    const int mt = wave / NT, nt = wave % NT;
    const int m0 = mt * 16, n0 = nt * 16;
    const int half = lane >> 4, l15 = lane & 15;
    v8f c = {};
    for (int kk = 0; kk < 16; ++kk) {
        const int k = kk * 4 + half * 2;
        v2f a, b;
        const float* ap = mat + (m0 + l15) * 64 + k;
        a.x = ap[0]; a.y = ap[1];
        const float* bp = Wmat + (n0 + l15) * 64 + k;   // B[k][n] = W[n][k]
        b.x = bp[0]; b.y = bp[1];
        c = __builtin_amdgcn_wmma_f32_16x16x4_f32(false, a, false, b, (short)0, c, false, false);
    }
    const int mrow = m0 + 8 * half, ncol = n0 + l15;
#pragma unroll
    for (int r = 0; r < 8; ++r) h[(mrow + r) * 64 + ncol] = c[r];
}

// ---------------- self attention scores + global max ----------------
__global__ void self_att_kernel(const float* __restrict__ h, const float* __restrict__ a_self,
                                float* __restrict__ scores, unsigned* __restrict__ maxkey) {
    const int i = blockIdx.x * blockDim.x + threadIdx.x;
    float s = -3.0e38f;
    if (i < N_MATN) {
        float acc = 0.f;
        const float* hp = h + i * 64;
        for (int j = 0; j < 64; ++j) acc += hp[j] * (a_self[j] + a_self[64 + j]);
        s = lrelu(acc);
        scores[i] = s;
    }
    __shared__ float red[256];
    red[threadIdx.x] = s;
    __syncthreads();
    for (int off = 128; off > 0; off >>= 1) {
        if (threadIdx.x < (unsigned)off) red[threadIdx.x] = fmaxf(red[threadIdx.x], red[threadIdx.x + off]);
        __syncthreads();
    }
    if (threadIdx.x == 0) atomicMax(maxkey, fkey(red[0]));
}

// ------- edge GEMM pass 1: accumulate ops_by_edges . a_cross[64:] per edge -------
__global__ void edge_gemm_att_kernel(const float* __restrict__ ops, const float* __restrict__ eattr,
                                     const int* __restrict__ ei, const float* __restrict__ Wop,
                                     const float* __restrict__ a_cross, float* __restrict__ cross_raw) {
    const int wave = (blockIdx.x * blockDim.x + threadIdx.x) >> 5;
    const int lane = threadIdx.x & 31;
    const int NT = D_EMB / 16;                      // 4
    const int total = (N_EDGE / 16) * NT;           // 200000
    if (wave >= total) return;
    const int mt = wave / NT, nt = wave % NT;
    const int m0 = mt * 16, n0 = nt * 16;
    const int half = lane >> 4, l15 = lane & 15;
    const int e = m0 + l15;
    const int src = ei[e];                          // edge_index[0][e]
    v8f c = {};
    for (int kk = 0; kk < 16; ++kk) {
        const int k = kk * 4 + half * 2;
        v2f a, b;
        const float* ap = (k < D_OPF) ? (ops + src * D_OPF + k)
                                      : (eattr + e * 16 + (k - D_OPF));
        a.x = ap[0]; a.y = ap[1];
        const float* bp = Wop + (n0 + l15) * 64 + k;
        b.x = bp[0]; b.y = bp[1];
        c = __builtin_amdgcn_wmma_f32_16x16x4_f32(false, a, false, b, (short)0, c, false, false);
    }
    const float ac = a_cross[64 + n0 + l15];
#pragma unroll
    for (int r = 0; r < 8; ++r) {
        float p = c[r] * ac;
        p += __shfl_xor(p, 1, 32);
        p += __shfl_xor(p, 2, 32);
        p += __shfl_xor(p, 4, 32);
        p += __shfl_xor(p, 8, 32);                  // sum over N within half-wave
        if (l15 == 0) atomicAdd(&cross_raw[m0 + 8 * half + r], p);
    }
}

// ------- finalize cross scores: add h[dst].a_cross[:64], lrelu, global max -------
__global__ void cross_finalize_kernel(const float* __restrict__ h, const int* __restrict__ ei,
                                      const float* __restrict__ a_cross,
                                      float* __restrict__ cross, unsigned* __restrict__ maxkey) {
    const int e = blockIdx.x * blockDim.x + threadIdx.x;
    float s = -3.0e38f;
    if (e < N_EDGE) {
        const int dst = ei[N_EDGE + e];             // edge_index[1][e]
        float acc = cross[e];
        const float* hp = h + dst * 64;
        for (int j = 0; j < 64; ++j) acc += hp[j] * a_cross[j];
        s = lrelu(acc);
        cross[e] = s;
    }
    __shared__ float red[256];
    red[threadIdx.x] = s;
    __syncthreads();
    for (int off = 128; off > 0; off >>= 1) {
        if (threadIdx.x < (unsigned)off) red[threadIdx.x] = fmaxf(red[threadIdx.x], red[threadIdx.x + off]);
        __syncthreads();
    }
    if (threadIdx.x == 0) atomicMax(maxkey, fkey(red[0]));
}

// ---------------- global softmax denominator ----------------
__global__ void expsum_kernel(const float* __restrict__ scores, const unsigned* __restrict__ maxkey,
                              float* __restrict__ denom) {
    const int i = blockIdx.x * blockDim.x + threadIdx.x;
    const float m = fkey_inv(*maxkey);
    float v = 0.f;
    if (i < N_MATN + N_EDGE) v = expf(scores[i] - m);
    __shared__ float red[256];
    red[threadIdx.x] = v;
    __syncthreads();
    for (int off = 128; off > 0; off >>= 1) {
        if (threadIdx.x < (unsigned)off) red[threadIdx.x] += red[threadIdx.x + off];
        __syncthreads();
    }
    if (threadIdx.x == 0) atomicAdd(denom, red[0]);
}

// ------- edge GEMM pass 2: recompute tile, scatter norm_cross * ops_by_edges -------
__global__ void edge_gemm_scatter_kernel(const float* __restrict__ ops, const float* __restrict__ eattr,
                                         const int* __restrict__ ei, const float* __restrict__ Wop,
                                         const float* __restrict__ cross, const unsigned* __restrict__ maxkey,
                                         const float* __restrict__ denom, float* __restrict__ out) {
    const int wave = (blockIdx.x * blockDim.x + threadIdx.x) >> 5;
    const int lane = threadIdx.x & 31;
    const int NT = D_EMB / 16;
    const int total = (N_EDGE / 16) * NT;
    if (wave >= total) return;
    const int mt = wave / NT, nt = wave % NT;
    const int m0 = mt * 16, n0 = nt * 16;
    const int half = lane >> 4, l15 = lane & 15;
    const int e = m0 + l15;
    const int src = ei[e];
    v8f c = {};
    for (int kk = 0; kk < 16; ++kk) {
        const int k = kk * 4 + half * 2;
        v2f a, b;
        const float* ap = (k < D_OPF) ? (ops + src * D_OPF + k)
                                      : (eattr + e * 16 + (k - D_OPF));
        a.x = ap[0]; a.y = ap[1];
        const float* bp = Wop + (n0 + l15) * 64 + k;
        b.x = bp[0]; b.y = bp[1];
        c = __builtin_amdgcn_wmma_f32_16x16x4_f32(false, a, false, b, (short)0, c, false, false);
    }
    const float m = fkey_inv(*maxkey);
    const float inv = 1.0f / *denom;
    const int ncol = n0 + l15;
#pragma unroll
    for (int r = 0; r < 8; ++r) {
        const int er = m0 + 8 * half + r;
        const float w = expf(cross[er] - m) * inv;  // norm_cross[er]
        const int dst = ei[N_EDGE + er];
        atomicAdd(&out[dst * 64 + ncol], w * c[r]);
    }
}

// ---------------- epilogue: elu(norm_self * h + summed) ----------------
__global__ void final_kernel(const float* __restrict__ h, const float* __restrict__ scores_self,
                             const unsigned* __restrict__ maxkey, const float* __restrict__ denom,
                             float* __restrict__ out) {
    const int idx = blockIdx.x * blockDim.x + threadIdx.x;
    if (idx >= N_MATN * D_EMB) return;
    const int i = idx >> 6;
    const float m = fkey_inv(*maxkey);
    const float inv = 1.0f / *denom;
    const float ns = expf(scores_self[i] - m) * inv;
    const float x = ns * h[idx] + out[idx];
    out[idx] = x > 0.f ? x : (expf(x) - 1.f);
}

extern "C" void kernel_launch(void* const* d_in, const int* in_sizes, int n_in,
                              void* d_out, int out_size, void* d_ws, size_t ws_size,
                              hipStream_t stream) {
    const float* materials  = (const float*)d_in[0];
    const float* operations = (const float*)d_in[1];
    const float* edge_attr  = (const float*)d_in[2];
    const int*   edge_index = (const int*)d_in[3];
    const float* W_mat      = (const float*)d_in[4];
    const float* W_op       = (const float*)d_in[5];
    const float* a_self     = (const float*)d_in[6];
    const float* a_cross    = (const float*)d_in[7];
    float* out = (float*)d_out;

    // workspace layout (floats): h[N_MAT*64] | scores[N_MAT+E] | maxkey | denom
    float*    h            = (float*)d_ws;
    float*    scores       = h + N_MATN * D_EMB;
    float*    scores_cross = scores + N_MATN;
    unsigned* maxkey       = (unsigned*)(scores + N_MATN + N_EDGE);
    float*    denom        = (float*)(maxkey + 1);

    const int elems = N_MATN * D_EMB;                       // 3.2M (covers scores too)
    init_ws_kernel<<<(elems + 255) / 256, 256, 0, stream>>>(out, scores, maxkey, denom);

    {   // h GEMM: 3125 x 4 tiles, 8 waves / 256-thread block
        const int waves = (N_MATN / 16) * (D_EMB / 16);     // 12500
        h_gemm_kernel<<<(waves + 7) / 8, 256, 0, stream>>>(materials, W_mat, h);
    }
    self_att_kernel<<<(N_MATN + 255) / 256, 256, 0, stream>>>(h, a_self, scores, maxkey);

    const int ewaves  = (N_EDGE / 16) * (D_EMB / 16);       // 200000
    const int eblocks = ewaves / 8;                         // 25000 exact
    edge_gemm_att_kernel<<<eblocks, 256, 0, stream>>>(operations, edge_attr, edge_index,
                                                      W_op, a_cross, scores_cross);
    cross_finalize_kernel<<<(N_EDGE + 255) / 256, 256, 0, stream>>>(h, edge_index, a_cross,
                                                                    scores_cross, maxkey);
    expsum_kernel<<<(N_MATN + N_EDGE + 255) / 256, 256, 0, stream>>>(scores, maxkey, denom);
    edge_gemm_scatter_kernel<<<eblocks, 256, 0, stream>>>(operations, edge_attr, edge_index,
                                                          W_op, scores_cross, maxkey, denom, out);
    final_kernel<<<(elems + 255) / 256, 256, 0, stream>>>(h, scores, maxkey, denom, out);
}